// LoLa_50311246905652
// MI455X (gfx1250) — compile-verified
//
#include <hip/hip_runtime.h>

// LoLa layer for MI455X (gfx1250, wave32).
//
// Key algebraic rewrite: the (B,N,N) distances tensor is never formed.
//   weighted_d[b,n] = mass[b,n]*rowsum_w[n] + (Wd@mass)[n,b]
//                     + 2*(x0*(Wd@x0) + x1*(Wd@x1) + x2*(Wd@x2) - x3*(Wd@x3))[n,b]
// Everything reduces to nine (512x512)@(512x128) fp32 GEMMs run on the
// matrix pipe with V_WMMA_F32_16X16X4_F32 (full fp32, K=4 per instruction).

#define BATCH 128
#define NPART 512
#define CH_STRIDE (NPART * BATCH)   // 65536 floats per packed channel

typedef float v2f __attribute__((ext_vector_type(2)));
typedef float v8f __attribute__((ext_vector_type(8)));

#if __has_builtin(__builtin_amdgcn_wmma_f32_16x16x4_f32)
__device__ __forceinline__ v8f wmma_f32x4(v2f a, v2f b, v8f c) {
  // 8-arg pattern: (neg_a, A, neg_b, B, c_mod, C, reuse_a, reuse_b)
  return __builtin_amdgcn_wmma_f32_16x16x4_f32(false, a, false, b, (short)0, c,
                                               false, false);
}
#else
// Fallback only so the TU keeps compiling if the builtin signature differs;
// a wmma=0 histogram tells us to fix the intrinsic next round.
__device__ __forceinline__ v8f wmma_f32x4(v2f a, v2f b, v8f c) {
  c[0] = fmaf(a[0], b[0], c[0]);
  return c;
}
#endif

// ---------------------------------------------------------------------------
// Kernel 1: pack combvec into WMMA-B-friendly layout + masses/ptsq epilogues.
// Packed layout per channel: P[ch][(j>>1)*256 + b*2 + (j&1)]  (j = particle,
// b = batch). A WMMA B-fragment lane then needs V[k][n],V[k+1][n] with k even,
// which is one aligned b64 load.
// Channels 0..5 = combvec features, channel 6 = masses.
// ---------------------------------------------------------------------------
__global__ __launch_bounds__(512) void lola_prep(const float* __restrict__ comb,
                                                 float* __restrict__ P,
                                                 float* __restrict__ out) {
  const int b = blockIdx.x;    // 0..127
  const int j = threadIdx.x;   // 0..511
  const float* c = comb + ((size_t)b * NPART + j) * 6;
  const float c0 = c[0], c1 = c[1], c2 = c[2], c3 = c[3], c4 = c[4], c5 = c[5];
  const float mass = c3 * c3 - c0 * c0 - c1 * c1 - c2 * c2;  // metric (-,-,-,+)
  const float ptsq = c1 * c1 + c2 * c2;

  const int pidx = (j >> 1) * (2 * BATCH) + b * 2 + (j & 1);
  P[0 * CH_STRIDE + pidx] = c0;
  P[1 * CH_STRIDE + pidx] = c1;
  P[2 * CH_STRIDE + pidx] = c2;
  P[3 * CH_STRIDE + pidx] = c3;
  P[4 * CH_STRIDE + pidx] = c4;
  P[5 * CH_STRIDE + pidx] = c5;
  P[6 * CH_STRIDE + pidx] = mass;

  float* o = out + ((size_t)b * NPART + j) * 7;
  o[0] = mass;
  o[1] = ptsq;
}

// ---------------------------------------------------------------------------
// Kernel 2: rowsum of w_dist (512 rows, L2-resident, trivial).
// ---------------------------------------------------------------------------
__global__ __launch_bounds__(256) void lola_rowsum(const float* __restrict__ w,
                                                   float* __restrict__ rs) {
  const int n = blockIdx.x * 256 + threadIdx.x;  // 0..511
  const float* row = w + (size_t)n * NPART;
  float s = 0.f;
#pragma unroll 8
  for (int m = 0; m < NPART; ++m) s += row[m];
  rs[n] = s;
}

// ---------------------------------------------------------------------------
// Kernel 3: the nine GEMMs on the WMMA pipe.
// grid = (32 M-tiles, 5 groups); block = 256 threads = 8 waves; each wave
// owns one 16-wide batch-column tile (8 waves cover all 128 batch columns).
// Groups 0..3: single GEMM each (ener, pid, extra0, extra1).
// Group 4: five w_dist GEMMs fused — one A fragment feeds 5 WMMAs per k-step,
//          then the distance recombination runs in the register epilogue.
//
// Fragment layouts (wave32, f32 16x16x4):
//   A(16x4): lane l<16 holds row M=l, K={0,1}; lane l>=16 holds K={2,3}.
//   B(4x16): vgpr v, lane-half h hold row K=h*2+v, col N=lane&15.
//   C(16x16): vgpr r, lane-half h hold row M=h*8+r, col N=lane&15.
// ---------------------------------------------------------------------------
__global__ __launch_bounds__(256) void lola_gemm(
    const float* __restrict__ w_dist, const float* __restrict__ w_ener,
    const float* __restrict__ w_pid, const float* __restrict__ w_x0,
    const float* __restrict__ w_x1, const float* __restrict__ P,
    const float* __restrict__ rowsum, float* __restrict__ out) {
  const int grp     = blockIdx.y;
  const int mtile   = blockIdx.x;
  const int lane    = threadIdx.x & 31;
  const int wave    = threadIdx.x >> 5;
  const int lane_lo = lane & 15;
  const int half    = lane >> 4;
  const int mrow    = mtile * 16 + lane_lo;  // A-fragment row
  const int nbase   = wave * 16;             // batch-column tile base
  const int bcol    = nbase + lane_lo;       // this lane's output batch column

  if (grp < 4) {
    const float* W;
    const float* V;
    int ok;
    if (grp == 0)      { W = w_ener; V = P + 0 * CH_STRIDE; ok = 2; }
    else if (grp == 1) { W = w_pid;  V = P + 3 * CH_STRIDE; ok = 4; }
    else if (grp == 2) { W = w_x0;   V = P + 4 * CH_STRIDE; ok = 5; }
    else               { W = w_x1;   V = P + 5 * CH_STRIDE; ok = 6; }

    const float* arow = W + (size_t)mrow * NPART + half * 2;
    const float* bptr = V + half * (2 * BATCH) + bcol * 2;

    v8f acc = {0.f, 0.f, 0.f, 0.f, 0.f, 0.f, 0.f, 0.f};
#pragma unroll 4
    for (int kk = 0; kk < NPART / 4; ++kk) {
      v2f a = *(const v2f*)(arow + kk * 4);
      v2f b = *(const v2f*)(bptr + kk * 4 * BATCH);  // advance 2 packed k-rows
      acc = wmma_f32x4(a, b, acc);
    }

    float* o = out + ((size_t)bcol * NPART + mtile * 16 + half * 8) * 7 + ok;
#pragma unroll
    for (int r = 0; r < 8; ++r) o[r * 7] = acc[r];
  } else {
    // Fused w_dist group: channels {mass, x0, x1, x2, x3}.
    const float* arow = w_dist + (size_t)mrow * NPART + half * 2;
    const size_t bof = (size_t)half * (2 * BATCH) + bcol * 2;
    const float* bm = P + 6 * CH_STRIDE + bof;
    const float* b0 = P + 0 * CH_STRIDE + bof;
    const float* b1 = P + 1 * CH_STRIDE + bof;
    const float* b2 = P + 2 * CH_STRIDE + bof;
    const float* b3 = P + 3 * CH_STRIDE + bof;

    v8f am = {0.f, 0.f, 0.f, 0.f, 0.f, 0.f, 0.f, 0.f};
    v8f a0 = am, a1 = am, a2 = am, a3 = am;
#pragma unroll 2
    for (int kk = 0; kk < NPART / 4; ++kk) {
      v2f a = *(const v2f*)(arow + kk * 4);
      am = wmma_f32x4(a, *(const v2f*)(bm + kk * 4 * BATCH), am);
      a0 = wmma_f32x4(a, *(const v2f*)(b0 + kk * 4 * BATCH), a0);
      a1 = wmma_f32x4(a, *(const v2f*)(b1 + kk * 4 * BATCH), a1);
      a2 = wmma_f32x4(a, *(const v2f*)(b2 + kk * 4 * BATCH), a2);
      a3 = wmma_f32x4(a, *(const v2f*)(b3 + kk * 4 * BATCH), a3);
    }

#pragma unroll
    for (int r = 0; r < 8; ++r) {
      const int n = mtile * 16 + half * 8 + r;
      const int pi = (n >> 1) * (2 * BATCH) + bcol * 2 + (n & 1);
      const float x0 = P[0 * CH_STRIDE + pi];
      const float x1 = P[1 * CH_STRIDE + pi];
      const float x2 = P[2 * CH_STRIDE + pi];
      const float x3 = P[3 * CH_STRIDE + pi];
      const float mv = P[6 * CH_STRIDE + pi];
      const float d = fmaf(mv, rowsum[n], am[r]) +
                      2.f * (x0 * a0[r] + x1 * a1[r] + x2 * a2[r] - x3 * a3[r]);
      out[((size_t)bcol * NPART + n) * 7 + 3] = d;
    }
  }
}

// ---------------------------------------------------------------------------
// Launch. d_ws layout: P (7*65536 floats) | rowsum (512 floats) = ~1.76 MB.
// ---------------------------------------------------------------------------
extern "C" void kernel_launch(void* const* d_in, const int* in_sizes, int n_in,
                              void* d_out, int out_size, void* d_ws,
                              size_t ws_size, hipStream_t stream) {
  const float* comb   = (const float*)d_in[0];
  const float* w_dist = (const float*)d_in[1];
  const float* w_ener = (const float*)d_in[2];
  const float* w_pid  = (const float*)d_in[3];
  const float* w_x0   = (const float*)d_in[4];
  const float* w_x1   = (const float*)d_in[5];
  float* out = (float*)d_out;

  float* P  = (float*)d_ws;              // 7 * 65536 floats
  float* rs = P + 7 * CH_STRIDE;         // 512 floats

  lola_prep<<<dim3(BATCH), dim3(NPART), 0, stream>>>(comb, P, out);
  lola_rowsum<<<dim3(2), dim3(256), 0, stream>>>(w_dist, rs);
  lola_gemm<<<dim3(NPART / 16, 5), dim3(256), 0, stream>>>(
      w_dist, w_ener, w_pid, w_x0, w_x1, P, rs, out);
}